// GraphDecoderHomo_16432544874897
// MI455X (gfx1250) — compile-verified
//
#include <hip/hip_runtime.h>
#include <hip/hip_bf16.h>

typedef __attribute__((ext_vector_type(2))) float v2f;
typedef __attribute__((ext_vector_type(8))) float v8f;

#define GCN_N 102400
#define GCN_E 1638400

// ---------------------------------------------------------------- utilities
__global__ void k_zero(float* __restrict__ p, int n) {
    int i = blockIdx.x * blockDim.x + threadIdx.x;
    if (i < n) p[i] = 0.0f;
}

__global__ void k_degree(const int* __restrict__ src, const int* __restrict__ dst,
                         float* __restrict__ odeg, float* __restrict__ ideg, int E) {
    int e = blockIdx.x * blockDim.x + threadIdx.x;
    if (e < E) {
        atomicAdd(&odeg[src[e]], 1.0f);
        atomicAdd(&ideg[dst[e]], 1.0f);
    }
}

// in-place: p[i] = (max(p[i],1))^-0.5
__global__ void k_norm(float* __restrict__ p, int n) {
    int i = blockIdx.x * blockDim.x + threadIdx.x;
    if (i < n) p[i] = rsqrtf(fmaxf(p[i], 1.0f));
}

// -------------------------------------------------- WMMA f32 GEMM (exact fp32)
// T[N,OUT] = (H row-scaled by onorm) @ W[IN,OUT]
// One wave per 16x16 output tile; block = (OUT/16) waves covering 16 rows.
// V_WMMA_F32_16X16X4_F32 operand layout (ISA 7.12.2):
//   A 16x4 : lane&15 = M, lane>>4 selects K-pair {0,1} vs {2,3} (VGPR0=K even slot)
//   B 4x16 : lane&15 = N, lane>>4 selects rows {0,1} vs {2,3}
//   D      : VGPR j -> row (half*8 + j), col lane&15
template <int IN, int OUT>
__global__ void k_gemm_wmma(const float* __restrict__ H, const float* __restrict__ onorm,
                            const float* __restrict__ W, float* __restrict__ T) {
    const int wave = threadIdx.x >> 5;
    const int lane = threadIdx.x & 31;
    const int half = lane >> 4;
    const int l16  = lane & 15;
    const int row0 = blockIdx.x * 16;
    const int col0 = wave * 16;

    const float* aRow  = H + (size_t)(row0 + l16) * IN;
    const float  scale = onorm[row0 + l16];

    v8f c = {};
    const int kOff = half * 2;
#pragma unroll 4
    for (int k = 0; k < IN; k += 4) {
        v2f a, b;
        a.x = aRow[k + kOff]     * scale;
        a.y = aRow[k + kOff + 1] * scale;
        b.x = W[(size_t)(k + kOff)     * OUT + col0 + l16];
        b.y = W[(size_t)(k + kOff + 1) * OUT + col0 + l16];
        c = __builtin_amdgcn_wmma_f32_16x16x4_f32(
                /*neg_a=*/false, a, /*neg_b=*/false, b,
                /*c_mod=*/(short)0, c, /*reuse_a=*/false, /*reuse_b=*/false);
    }

    float* out = T + (size_t)(row0 + half * 8) * OUT + col0 + l16;
#pragma unroll
    for (int j = 0; j < 8; ++j)
        out[(size_t)j * OUT] = c[j];
}

// ---------------------------------------------- edge gather * ew -> scatter-add
// one (edge, feature) pair per thread: coalesced gather + coalesced atomics
template <int F>
__global__ void k_edge(const int* __restrict__ src, const int* __restrict__ dst,
                       const float* __restrict__ ew, const float* __restrict__ T,
                       float* __restrict__ agg, int E) {
    unsigned idx = blockIdx.x * blockDim.x + threadIdx.x;
    if (idx >= (unsigned)E * F) return;
    unsigned e = idx / F;
    unsigned f = idx - e * F;
    float v = T[(size_t)src[e] * F + f] * ew[e];
    atomicAdd(&agg[(size_t)dst[e] * F + f], v);
}

// agg = [tanh](agg * inorm[row] + bias[col]) in place
template <bool TANH>
__global__ void k_post(float* __restrict__ agg, const float* __restrict__ inorm,
                       const float* __restrict__ bias, int F, int N) {
    unsigned idx = blockIdx.x * blockDim.x + threadIdx.x;
    if (idx >= (unsigned)N * F) return;
    unsigned n = idx / F;
    unsigned f = idx - n * F;
    float v = agg[idx] * inorm[n] + bias[f];
    agg[idx] = TANH ? tanhf(v) : v;
}

// final layer transform: T[n] = onorm[n] * dot(H[n,0:32], W[0:32])
__global__ void k_dot32(const float* __restrict__ H, const float* __restrict__ onorm,
                        const float* __restrict__ W, float* __restrict__ T, int N) {
    int n = blockIdx.x * blockDim.x + threadIdx.x;
    if (n >= N) return;
    const float* h = H + (size_t)n * 32;
    float s = 0.0f;
#pragma unroll
    for (int i = 0; i < 32; ++i) s += h[i] * W[i];
    T[n] = s * onorm[n];
}

// ---------------------------------------------------------------- launcher
extern "C" void kernel_launch(void* const* d_in, const int* in_sizes, int n_in,
                              void* d_out, int out_size, void* d_ws, size_t ws_size,
                              hipStream_t stream) {
    const float* b_z = (const float*)d_in[0];   // [N,128]
    const int*   src = (const int*)d_in[1];     // [E]
    const int*   dst = (const int*)d_in[2];     // [E]
    const float* ew  = (const float*)d_in[3];   // [E]
    // d_in[4] = b_size scalar (unused; out is the flat node vector)
    const float* W0 = (const float*)d_in[5];    // [128,64]
    const float* b0 = (const float*)d_in[6];    // [64]
    const float* W1 = (const float*)d_in[7];    // [64,32]
    const float* b1 = (const float*)d_in[8];    // [32]
    const float* W2 = (const float*)d_in[9];    // [32,1]
    const float* b2 = (const float*)d_in[10];   // [1]

    const int N = GCN_N, E = GCN_E;

    // workspace layout (floats): onorm[N] | inorm[N] | bufA[N*64] | bufB[N*64]
    float* ws    = (float*)d_ws;
    float* onorm = ws;
    float* inorm = ws + (size_t)N;
    float* bufA  = ws + (size_t)2 * N;
    float* bufB  = bufA + (size_t)N * 64;
    float* outF  = (float*)d_out;

    const int TB = 256;
    auto blocks = [](long long n, int tb) { return (unsigned)((n + tb - 1) / tb); };

    // --- degrees -> symmetric norms (shared across all 3 layers) ---
    k_zero<<<blocks(2LL * N, TB), TB, 0, stream>>>(onorm, 2 * N);
    k_degree<<<blocks(E, TB), TB, 0, stream>>>(src, dst, onorm, inorm, E);
    k_norm<<<blocks(2LL * N, TB), TB, 0, stream>>>(onorm, 2 * N);

    // --- layer 0: 128 -> 64, tanh ---
    k_gemm_wmma<128, 64><<<N / 16, 128, 0, stream>>>(b_z, onorm, W0, bufA);
    k_zero<<<blocks((long long)N * 64, TB), TB, 0, stream>>>(bufB, N * 64);
    k_edge<64><<<blocks((long long)E * 64, TB), TB, 0, stream>>>(src, dst, ew, bufA, bufB, E);
    k_post<true><<<blocks((long long)N * 64, TB), TB, 0, stream>>>(bufB, inorm, b0, 64, N);

    // --- layer 1: 64 -> 32, tanh ---
    k_gemm_wmma<64, 32><<<N / 16, 64, 0, stream>>>(bufB, onorm, W1, bufA);
    k_zero<<<blocks((long long)N * 32, TB), TB, 0, stream>>>(bufB, N * 32);
    k_edge<32><<<blocks((long long)E * 32, TB), TB, 0, stream>>>(src, dst, ew, bufA, bufB, E);
    k_post<true><<<blocks((long long)N * 32, TB), TB, 0, stream>>>(bufB, inorm, b1, 32, N);

    // --- layer 2: 32 -> 1, no activation; aggregate straight into d_out ---
    k_dot32<<<blocks(N, TB), TB, 0, stream>>>(bufB, onorm, W2, bufA, N);
    k_zero<<<blocks(N, TB), TB, 0, stream>>>(outF, N);
    k_edge<1><<<blocks(E, TB), TB, 0, stream>>>(src, dst, ew, bufA, outF, E);
    k_post<false><<<blocks(N, TB), TB, 0, stream>>>(outF, inorm, b2, 1, N);
}